// MultitaskAgent_38233798869715
// MI455X (gfx1250) — compile-verified
//
#include <hip/hip_runtime.h>
#include <hip/hip_bf16.h>
#include <math.h>

typedef __attribute__((ext_vector_type(2))) float v2f;
typedef __attribute__((ext_vector_type(8))) float v8f;
typedef int aiv4 __attribute__((vector_size(16)));  // async builtin's 16B pointee

#if __has_builtin(__builtin_amdgcn_global_load_async_to_lds_b128) && \
    __has_builtin(__builtin_amdgcn_s_wait_asynccnt)
#define USE_ASYNC_LDS 1
#else
#define USE_ASYNC_LDS 0
#endif

#define AS3 __attribute__((address_space(3)))

// ---------------------------------------------------------------------------
// f32 WMMA 16x16 tile GEMM helper: C += A(16xK) * B(Kx16)
// A row-major [m][k] stride lda; B given transposed as Bt[n][k] stride ldb.
// ISA 7.12.2 f32 layouts: lane r(0..15)/hf(0..1):
//   A VGPR0 = A[r][k0+2*hf], VGPR1 = A[r][k0+2*hf+1]
//   B VGPR0 = B[k0+2*hf][n=r] = Bt[r][k0+2*hf]
//   C/D vgpr i -> row (i + 8*hf), col r
// ---------------------------------------------------------------------------
__device__ __forceinline__ v8f wgemm_tile(const float* __restrict__ A, int lda,
                                          const float* __restrict__ Bt, int ldb,
                                          int K, int r, int hf, v8f acc) {
  const float* ar = A + r * lda + 2 * hf;
  const float* br = Bt + r * ldb + 2 * hf;
  for (int k0 = 0; k0 < K; k0 += 4) {
    v2f av, bv;
    av[0] = ar[k0];
    av[1] = ar[k0 + 1];
    bv[0] = br[k0];
    bv[1] = br[k0 + 1];
    acc = __builtin_amdgcn_wmma_f32_16x16x4_f32(false, av, false, bv,
                                                (short)0, acc, false, false);
  }
  return acc;
}

__device__ __forceinline__ float lrelu(float v) { return (v >= 0.f) ? v : 0.01f * v; }

// Sizes
#define BS 128
#define NE 256
#define NO 8
#define HH 32
#define KK 16
#define ID 16
#define IC 16
#define NI 10
#define IH 4
#define OH 2
#define NA 531     // senders
#define NAP 544    // padded senders (34 * 16)
#define NR 1032    // receivers
#define NRP 1056   // padded receivers (33 * 32)

#define ZLD 50
#define WLD 49
#define K1LD 34

// ===========================================================================
// Kernel A: build sender_keys / sender_vals / recv_q (one block per batch)
// ===========================================================================
__global__ void __launch_bounds__(256) agent_kA(
    const float* __restrict__ x, const float* __restrict__ err,
    const float* __restrict__ oaddr, const float* __restrict__ hid,
    const float* __restrict__ sid,
    const float* __restrict__ k1w, const float* __restrict__ k1b,
    const float* __restrict__ k2w, const float* __restrict__ k2b,
    const float* __restrict__ vw, const float* __restrict__ vb,
    const float* __restrict__ qw, const float* __restrict__ qb,
    const float* __restrict__ ek, const float* __restrict__ evw,
    const float* __restrict__ evb,
    const float* __restrict__ ikw, const float* __restrict__ ikb,
    const float* __restrict__ ivw, const float* __restrict__ ivb,
    const float* __restrict__ stub,
    float* __restrict__ skeys, float* __restrict__ svals,
    float* __restrict__ rqo) {
  __shared__ float zs[NE * ZLD];      // z: 256 x 48 (padded)
  __shared__ float wsb[32 * WLD];     // weight tile (max 32 x 48)
  __shared__ float k1buf[NE * K1LD];  // k1: 256 x 32 (padded)

  const int b = blockIdx.x;
  const int tid = threadIdx.x;
  const int wave = __builtin_amdgcn_readfirstlane(tid >> 5);  // wave-uniform
  const int lane = tid & 31;
  const int hf = lane >> 4;
  const int r = lane & 15;

  float* SK = skeys + (size_t)b * NAP * KK;
  float* SV = svals + (size_t)b * NAP * HH;
  float* RQ = rqo + (size_t)b * NRP * KK;

  // ---- small senders / receivers (scalar; tiny) ----
  for (int i = tid; i < NI * KK; i += 256) {  // ik_in rows 0..9
    int row = i >> 4, k = i & 15;
    float acc = ikb[k];
    for (int c = 0; c < IC; ++c) acc += x[b * IC * NI + c * NI + row] * ikw[k * IC + c];
    SK[row * KK + k] = acc;
  }
  for (int i = tid; i < NI * HH; i += 256) {  // iv_in rows 0..9
    int row = i >> 5, v = i & 31;
    float acc = ivb[v];
    for (int c = 0; c < IC; ++c) acc += x[b * IC * NI + c * NI + row] * ivw[v * IC + c];
    SV[row * HH + v] = acc;
  }
  for (int i = tid; i < NO * KK; i += 256) {  // out_qs -> recv_q rows 0..7
    int j = i >> 4, k = i & 15;
    RQ[j * KK + k] = (k < 8) ? stub[k] : oaddr[b * 64 + j * 8 + (k - 8)];
  }
  for (int i = tid; i < NO * KK; i += 256) {  // ik_out rows 10..17
    int j = i >> 4, k = i & 15;
    float acc = ikb[k];
    for (int c = 0; c < IC; ++c) {
      float q = (c < 8) ? stub[c] : oaddr[b * 64 + j * 8 + (c - 8)];
      acc += q * ikw[k * IC + c];
    }
    SK[(10 + j) * KK + k] = acc;
  }
  for (int i = tid; i < NO * HH; i += 256) {  // iv_out rows 10..17
    int j = i >> 5, v = i & 31;
    float acc = ivb[v];
    for (int c = 0; c < IC; ++c) {
      float q = (c < 8) ? stub[c] : oaddr[b * 64 + j * 8 + (c - 8)];
      acc += q * ivw[v * IC + c];
    }
    SV[(10 + j) * HH + v] = acc;
  }
  if (tid < KK) SK[18 * KK + tid] = ek[tid];                       // errk row 18
  if (tid < HH) SV[18 * HH + tid] = err[b] * evw[tid] + evb[tid];  // errv row 18
  // zero pad rows
  for (int i = tid; i < (NAP - NA) * KK; i += 256) SK[NA * KK + i] = 0.f;
  for (int i = tid; i < (NAP - NA) * HH; i += 256) SV[NA * HH + i] = 0.f;
  for (int i = tid; i < (NRP - NR) * KK; i += 256) RQ[NR * KK + i] = 0.f;

  // ---- stage z = [hid; static_id]^T (256 x 48) in LDS ----
  for (int i = tid; i < NE * 48; i += 256) {
    int e = i / 48, c = i % 48;
    zs[e * ZLD + c] =
        (c < HH) ? hid[b * HH * NE + c * NE + e] : sid[b * ID * NE + (c - HH) * NE + e];
  }
  __syncthreads();

  // ---- k1 -> ak per OH head ----
  for (int h = 0; h < OH; ++h) {
    for (int i = tid; i < HH * 48; i += 256)  // key1_w[h]: 32 x 48
      wsb[(i / 48) * WLD + (i % 48)] = k1w[(h * HH + i / 48) * 48 + (i % 48)];
    __syncthreads();
    for (int tt = wave; tt < 32; tt += 8) {  // M=256, N=32, K=48
      int a0 = (tt >> 1) * 16, n0 = (tt & 1) * 16;
      v8f acc = {};
      acc = wgemm_tile(zs + a0 * ZLD, ZLD, wsb + n0 * WLD, WLD, 48, r, hf, acc);
      float bias = k1b[h * HH + n0 + r];
      for (int i = 0; i < 8; ++i)
        k1buf[(a0 + i + 8 * hf) * K1LD + n0 + r] = lrelu(acc[i] + bias);
    }
    __syncthreads();
    for (int i = tid; i < KK * HH; i += 256)  // key2_w[h]: 16 x 32
      wsb[(i / 32) * WLD + (i % 32)] = k2w[(h * KK + i / 32) * 32 + (i % 32)];
    __syncthreads();
    for (int tt = wave; tt < 16; tt += 8) {  // ak: M=256, N=16, K=32
      int a0 = tt * 16;
      v8f acc = {};
      acc = wgemm_tile(k1buf + a0 * K1LD, K1LD, wsb, WLD, 32, r, hf, acc);
      float bias = k2b[h * KK + r];
      for (int i = 0; i < 8; ++i)
        SK[(19 + h * NE + a0 + i + 8 * hf) * KK + r] = acc[i] + bias;
    }
    __syncthreads();
  }

  // ---- aq per IH head: M=256, N=16, K=48 ----
  for (int h = 0; h < IH; ++h) {
    for (int i = tid; i < KK * 48; i += 256)
      wsb[(i / 48) * WLD + (i % 48)] = qw[(h * KK + i / 48) * 48 + (i % 48)];
    __syncthreads();
    for (int tt = wave; tt < 16; tt += 8) {
      int a0 = tt * 16;
      v8f acc = {};
      acc = wgemm_tile(zs + a0 * ZLD, ZLD, wsb, WLD, 48, r, hf, acc);
      float bias = qb[h * KK + r];
      for (int i = 0; i < 8; ++i)
        RQ[(NO + h * NE + a0 + i + 8 * hf) * KK + r] = acc[i] + bias;
    }
    __syncthreads();
  }

  // ---- av per OH head: M=256, N=32, K=48 ----
  for (int h = 0; h < OH; ++h) {
    for (int i = tid; i < HH * 48; i += 256)
      wsb[(i / 48) * WLD + (i % 48)] = vw[(h * HH + i / 48) * 48 + (i % 48)];
    __syncthreads();
    for (int tt = wave; tt < 32; tt += 8) {
      int a0 = (tt >> 1) * 16, n0 = (tt & 1) * 16;
      v8f acc = {};
      acc = wgemm_tile(zs + a0 * ZLD, ZLD, wsb + n0 * WLD, WLD, 48, r, hf, acc);
      float bias = vb[h * HH + n0 + r];
      for (int i = 0; i < 8; ++i)
        SV[(19 + h * NE + a0 + i + 8 * hf) * HH + n0 + r] = acc[i] + bias;
    }
    __syncthreads();
  }
}

// ===========================================================================
// Kernel B: scores -> softmax -> w (output) -> vals -> row-norm
// grid: (33 receiver tiles, 128 batches)
// ===========================================================================
#define SKLD 20  // 80B rows: 16B-aligned for async B128, conflict-free (period-16 bank walk)
#define SCLD 33

__global__ void __launch_bounds__(256) agent_kB(
    const float* __restrict__ skeys, const float* __restrict__ svals,
    const float* __restrict__ rqg, float* __restrict__ outw,
    float* __restrict__ valsout) {
  __shared__ float sk[NAP * SKLD];   // 544 x 16 sender keys
  __shared__ float rq[32 * SKLD];    // 32 x 16 receiver queries
  __shared__ float sc[NAP * SCLD];   // 544 x 32 scores -> probabilities
  __shared__ float red[8 * 32];
  __shared__ float colmax[32];
  __shared__ float colsum[32];
  __shared__ float valbuf0[32 * SCLD];  // split-K partials (k in [0,272))
  __shared__ float valbuf1[32 * SCLD];  // split-K partials (k in [272,544))

  const int t = blockIdx.x;  // receiver tile
  const int b = blockIdx.y;
  const int tid = threadIdx.x;
  const int wave = __builtin_amdgcn_readfirstlane(tid >> 5);  // wave-uniform
  const int lane = tid & 31;
  const int hf = lane >> 4;
  const int r = lane & 15;

  const float* SKg = skeys + (size_t)b * NAP * KK;
  const float* SVg = svals + (size_t)b * NAP * HH;
  const float* RQg = rqg + (size_t)b * NRP * KK;

#if USE_ASYNC_LDS
  // Async global->LDS staging (GLOBAL_LOAD_ASYNC_TO_LDS_B128, ASYNCcnt).
  // Each 16B chunk stays inside one 80B (16B-aligned) padded LDS row.
  for (int i = tid; i < NAP * 4; i += 256) {  // 544 rows x 4 chunks
    int row = i >> 2, c4 = i & 3;
    __builtin_amdgcn_global_load_async_to_lds_b128(
        (aiv4*)((const char*)SKg + (size_t)i * 16),
        (AS3 aiv4*)((char*)sk + ((size_t)row * SKLD + c4 * 4) * 4), 0, 0);
  }
  if (tid < 32 * 4) {  // rq tile: 32 rows x 4 chunks
    int row = tid >> 2, c4 = tid & 3;
    __builtin_amdgcn_global_load_async_to_lds_b128(
        (aiv4*)((const char*)RQg + ((size_t)t * 128 + tid) * 16),
        (AS3 aiv4*)((char*)rq + ((size_t)row * SKLD + c4 * 4) * 4), 0, 0);
  }
  __builtin_amdgcn_s_wait_asynccnt(0);
#else
  for (int i = tid; i < NAP * KK; i += 256) sk[(i >> 4) * SKLD + (i & 15)] = SKg[i];
  for (int i = tid; i < 32 * KK; i += 256) {
    int n = i >> 4, k = i & 15;
    rq[n * SKLD + k] = RQg[(t * 32 + n) * KK + k];
  }
#endif
  __syncthreads();

  // scores = (keys . queries) / sqrt(16) * beta = dot * 0.625
  for (int tt = wave; tt < 68; tt += 8) {  // 34 a-tiles x 2 n-tiles
    int a0 = (tt >> 1) * 16, n0 = (tt & 1) * 16;
    v8f acc = {};
    acc = wgemm_tile(sk + a0 * SKLD, SKLD, rq + n0 * SKLD, SKLD, 16, r, hf, acc);
    for (int i = 0; i < 8; ++i)
      sc[(a0 + i + 8 * hf) * SCLD + n0 + r] = acc[i] * 0.625f;
  }
  __syncthreads();

  // stable softmax down the sender axis (531 valid rows)
  const int g = wave;
  const int n = tid & 31;
  const int aStart = g * 68, aEnd = aStart + 68;
  float m = -1e30f;
  for (int a = aStart; a < aEnd; ++a)
    if (a < NA) m = fmaxf(m, sc[a * SCLD + n]);
  red[g * 32 + n] = m;
  __syncthreads();
  if (tid < 32) {
    float mm = red[tid];
    for (int gg = 1; gg < 8; ++gg) mm = fmaxf(mm, red[gg * 32 + tid]);
    colmax[tid] = mm;
  }
  __syncthreads();
  float cm = colmax[n];
  float s = 0.f;
  for (int a = aStart; a < aEnd; ++a) {
    if (a < NA) {
      float e = __expf(sc[a * SCLD + n] - cm);
      sc[a * SCLD + n] = e;
      s += e;
    } else {
      sc[a * SCLD + n] = 0.f;  // padded senders contribute nothing
    }
  }
  red[g * 32 + n] = s;
  __syncthreads();
  if (tid < 32) {
    float ss = red[tid];
    for (int gg = 1; gg < 8; ++gg) ss += red[gg * 32 + tid];
    colsum[tid] = ss;
  }
  __syncthreads();
  float inv = 1.f / colsum[n];
  int gn = t * 32 + n;
  for (int a = aStart; a < aEnd; ++a) {
    if (a < NA) {
      float w = sc[a * SCLD + n] * inv;
      sc[a * SCLD + n] = w;
      if (gn < NR) outw[(size_t)b * NA * NR + (size_t)a * NR + gn] = w;
    }
  }
  __syncthreads();

  // vals(32 x 32) = w^T (32 x 531) @ sender_vals (531 x 32)
  // split-K across all 8 waves: wave = {tile 0..3} x {k-half 0..1}
  {
    int ct = wave & 3, kh = wave >> 2;
    int m0 = (ct & 1) * 16, v0 = (ct >> 1) * 16;
    int kBase = kh * (NAP / 2);
    v8f acc = {};
    const float* scp = sc + (kBase + 2 * hf) * SCLD + m0 + r;  // A[m][k] = sc[k][m0+m]
    const float* svp = SVg + (kBase + 2 * hf) * HH + v0 + r;   // B[k][v]
    for (int k0 = 0; k0 < NAP / 2; k0 += 4) {
      v2f av, bv;
      av[0] = scp[k0 * SCLD];
      av[1] = scp[k0 * SCLD + SCLD];
      bv[0] = svp[k0 * HH];
      bv[1] = svp[k0 * HH + HH];
      acc = __builtin_amdgcn_wmma_f32_16x16x4_f32(false, av, false, bv,
                                                  (short)0, acc, false, false);
    }
    float* pb = kh ? valbuf1 : valbuf0;
    for (int i = 0; i < 8; ++i)
      pb[(m0 + i + 8 * hf) * SCLD + v0 + r] = acc[i];
  }
  __syncthreads();

  // row normalize (mean/std over v=32, ddof=1) via wave32 shuffles
  for (int rr = wave; rr < 32; rr += 8) {
    float v = valbuf0[rr * SCLD + lane] + valbuf1[rr * SCLD + lane];
    float sum = v;
    for (int off = 16; off > 0; off >>= 1) sum += __shfl_xor(sum, off, 32);
    float mu = sum * (1.f / 32.f);
    float d = v - mu;
    float ss = d * d;
    for (int off = 16; off > 0; off >>= 1) ss += __shfl_xor(ss, off, 32);
    float sd = sqrtf(ss * (1.f / 31.f));
    float o = d / (sd + 1e-8f);
    int grow = t * 32 + rr;
    if (grow < NR) valsout[((size_t)b * NR + grow) * HH + lane] = o;
  }
}

// ===========================================================================
// Kernel C: vn1 -> vn2 -> channel norm -> gated hidden update
// grid: (4 column tiles of 64, 128 batches)
// ===========================================================================
#define Z2LD 65
__global__ void __launch_bounds__(256) agent_kC(
    const float* __restrict__ vals, const float* __restrict__ hid,
    const float* __restrict__ sid,
    const float* __restrict__ vn1w, const float* __restrict__ vn1b,
    const float* __restrict__ vn2w, const float* __restrict__ vn2b,
    const float* __restrict__ htohw, const float* __restrict__ htohb,
    const float* __restrict__ htogw, const float* __restrict__ htogb,
    float* __restrict__ newhid) {
  __shared__ float z2[176 * Z2LD];
  __shared__ float z2a[32 * Z2LD];
  __shared__ float z2b[32 * Z2LD];
  const int b = blockIdx.y;
  const int e0 = blockIdx.x * 64;
  const int tid = threadIdx.x;

  for (int i = tid; i < 176 * 64; i += 256) {
    int c = i >> 6, e = i & 63;
    int ge = e0 + e;
    float v;
    if (c < 128) {  // res channel c = ih*32 + vv -> vals[8 + ih*256 + e][vv]
      int ih = c >> 5, vv = c & 31;
      v = vals[((size_t)b * NR + (NO + ih * NE + ge)) * HH + vv];
    } else if (c < 160) {
      v = hid[b * HH * NE + (c - 128) * NE + ge];
    } else {
      v = sid[b * ID * NE + (c - 160) * NE + ge];
    }
    z2[c * Z2LD + e] = v;
  }
  __syncthreads();
  const int e = tid & 63;
  const int ob = (tid >> 6) * 8;
  for (int o = ob; o < ob + 8; ++o) {
    float acc = vn1b[o];
    for (int c = 0; c < 176; ++c) acc += vn1w[o * 176 + c] * z2[c * Z2LD + e];
    z2a[o * Z2LD + e] = lrelu(acc);
  }
  __syncthreads();
  for (int o = ob; o < ob + 8; ++o) {
    float acc = vn2b[o];
    for (int c = 0; c < 32; ++c) acc += vn2w[o * 32 + c] * z2a[c * Z2LD + e];
    z2b[o * Z2LD + e] = lrelu(acc);
  }
  __syncthreads();
  if (tid < 64) {  // channel-norm (32 channels per column, ddof=1)
    float sum = 0.f;
    for (int o = 0; o < 32; ++o) sum += z2b[o * Z2LD + tid];
    float mu = sum / 32.f;
    float ss = 0.f;
    for (int o = 0; o < 32; ++o) {
      float d = z2b[o * Z2LD + tid] - mu;
      ss += d * d;
    }
    float inv = 1.f / (sqrtf(ss / 31.f) + 1e-8f);
    for (int o = 0; o < 32; ++o)
      z2b[o * Z2LD + tid] = (z2b[o * Z2LD + tid] - mu) * inv;
  }
  __syncthreads();
  for (int o = ob; o < ob + 8; ++o) {
    float ag = htogb[o], ah = htohb[o];
    for (int c = 0; c < 32; ++c) {
      float zv = z2b[c * Z2LD + e];
      ag += htogw[o * 64 + c] * zv;
      ah += htohw[o * 64 + c] * zv;
    }
    for (int c = 0; c < 32; ++c) {
      float hv = z2[(128 + c) * Z2LD + e];
      ag += htogw[o * 64 + 32 + c] * hv;
      ah += htohw[o * 64 + 32 + c] * hv;
    }
    float gg = 1.f / (1.f + __expf(-ag));
    float hv = z2[(128 + o) * Z2LD + e];
    newhid[b * HH * NE + o * NE + e0 + e] = ah * gg + hv * (1.f - gg);
  }
}

// ===========================================================================
// Kernel D: output head (128 x 8)
// ===========================================================================
__global__ void agent_kD(const float* __restrict__ vals,
                         const float* __restrict__ ovw,
                         const float* __restrict__ ovb,
                         float* __restrict__ out) {
  int i = blockIdx.x * blockDim.x + threadIdx.x;
  if (i >= BS * NO) return;
  int b = i >> 3, nn = i & 7;
  float acc = ovb[0];
  for (int v = 0; v < HH; ++v)
    acc += vals[((size_t)b * NR + nn) * HH + v] * ovw[v];
  out[i] = acc;
}

extern "C" void kernel_launch(void* const* d_in, const int* in_sizes, int n_in,
                              void* d_out, int out_size, void* d_ws, size_t ws_size,
                              hipStream_t stream) {
  (void)in_sizes; (void)n_in; (void)out_size; (void)ws_size;
  const float* x = (const float*)d_in[0];
  const float* err = (const float*)d_in[1];
  const float* oaddr = (const float*)d_in[2];
  const float* hid = (const float*)d_in[3];
  const float* sid = (const float*)d_in[4];
  const float* k1w = (const float*)d_in[5];
  const float* k1b = (const float*)d_in[6];
  const float* k2w = (const float*)d_in[7];
  const float* k2b = (const float*)d_in[8];
  const float* vw = (const float*)d_in[9];
  const float* vb = (const float*)d_in[10];
  const float* qw = (const float*)d_in[11];
  const float* qb = (const float*)d_in[12];
  const float* vn1w = (const float*)d_in[13];
  const float* vn1b = (const float*)d_in[14];
  const float* vn2w = (const float*)d_in[15];
  const float* vn2b = (const float*)d_in[16];
  const float* htohw = (const float*)d_in[17];
  const float* htohb = (const float*)d_in[18];
  const float* htogw = (const float*)d_in[19];
  const float* htogb = (const float*)d_in[20];
  const float* ek = (const float*)d_in[21];
  const float* evw = (const float*)d_in[22];
  const float* evb = (const float*)d_in[23];
  const float* ikw = (const float*)d_in[24];
  const float* ikb = (const float*)d_in[25];
  const float* ivw = (const float*)d_in[26];
  const float* ivb = (const float*)d_in[27];
  const float* stub = (const float*)d_in[28];
  const float* ovw = (const float*)d_in[29];
  const float* ovb = (const float*)d_in[30];

  float* ws = (float*)d_ws;
  float* skeys = ws;                                    // 128*544*16
  float* svals = skeys + (size_t)BS * NAP * KK;         // 128*544*32
  float* rq = svals + (size_t)BS * NAP * HH;            // 128*1056*16
  float* vals = rq + (size_t)BS * NRP * KK;             // 128*1032*32

  float* out = (float*)d_out;                           // (128,8)
  float* outw = out + BS * NO;                          // (128,531,1032)
  float* newhid = out + (size_t)BS * NO + (size_t)BS * NA * NR;  // (128,32,256)

  agent_kA<<<BS, 256, 0, stream>>>(x, err, oaddr, hid, sid, k1w, k1b, k2w, k2b,
                                   vw, vb, qw, qb, ek, evw, evb, ikw, ikb, ivw,
                                   ivb, stub, skeys, svals, rq);
  agent_kB<<<dim3(33, BS), 256, 0, stream>>>(skeys, svals, rq, outw, vals);
  agent_kC<<<dim3(4, BS), 256, 0, stream>>>(vals, hid, sid, vn1w, vn1b, vn2w,
                                            vn2b, htohw, htohb, htogw, htogb,
                                            newhid);
  agent_kD<<<4, 256, 0, stream>>>(vals, ovw, ovb, out);
}